// ProteinBackboneFeature_28312424415534
// MI455X (gfx1250) — compile-verified
//
#include <hip/hip_runtime.h>

typedef __attribute__((ext_vector_type(2))) float v2f;
typedef __attribute__((ext_vector_type(8))) float v8f;

namespace {
constexpr int NB   = 50;                // beads
constexpr int NBP  = 64;                // padded beads (4 WMMA tiles)
constexpr int ND   = 1225;              // pair distances
constexpr int NA   = 48;                // angles
constexpr int NDH  = 47;                // dihedrals
constexpr int NF   = ND + NA + 2 * NDH; // 1367 features per frame
constexpr int GP   = 65;                // Gram LDS row pitch (bank-conflict pad)
constexpr int BLK  = 320;               // 10 wave32 = one WMMA tile pair per wave
// packed 2-bit (ti,tj) tables for the 10 upper-triangle 16x16 tile pairs:
// p: 0:(0,0) 1:(0,1) 2:(0,2) 3:(0,3) 4:(1,1) 5:(1,2) 6:(1,3) 7:(2,2) 8:(2,3) 9:(3,3)
constexpr unsigned TI_PACK = 0xE9500u;
constexpr unsigned TJ_PACK = 0xFB9E4u;

__device__ __forceinline__ float fast_sqrt(float x) {
#if __has_builtin(__builtin_amdgcn_sqrtf)
    return __builtin_amdgcn_sqrtf(x);   // bare v_sqrt_f32 (~1 ulp), no fixup chain
#else
    return sqrtf(x);
#endif
}
}

__global__ __launch_bounds__(BLK) void backbone_feat_kernel(
    const float* __restrict__ data, float* __restrict__ out, int n_frames)
{
    __shared__ float sx[NBP], sy[NBP], sz[NBP]; // SoA coords, zero-padded tail
    __shared__ float G[64 * GP];                // padded Gram matrix

    const int frame = blockIdx.x;
    if (frame >= n_frames) return;
    const int tid = threadIdx.x;

    // ---- Stage frame coords (AoS -> zero-padded SoA in LDS) ----
    const float* src = data + (size_t)frame * (NB * 3);
    if (tid < NB * 3) {
        const float v = src[tid];
        const int bead = tid / 3, comp = tid - 3 * bead;
        float* dst = (comp == 0) ? sx : (comp == 1) ? sy : sz;
        dst[bead] = v;
    } else if (tid < NB * 3 + 3 * (NBP - NB)) {     // zero the 42 pad slots
        const int q = tid - NB * 3;
        const int comp = q / (NBP - NB), bead = NB + (q - comp * (NBP - NB));
        float* dst = (comp == 0) ? sx : (comp == 1) ? sy : sz;
        dst[bead] = 0.f;
    }
    __syncthreads();

    const int lane = tid & 31;
    const int wave = tid >> 5;        // 0..9 -> exactly one tile pair
    const int half = lane >> 4;       // lanes 16-31: K=2,3 (A/B) / M+8 (D)
    const int m    = lane & 15;

    // ---- Gram via V_WMMA_F32_16X16X4_F32, one 16x16 tile per wave ----
    {
        const int ti = (TI_PACK >> (2 * wave)) & 3;
        const int tj = (TJ_PACK >> (2 * wave)) & 3;
        const int r = ti * 16 + m;    // A row (M)
        const int n = tj * 16 + m;    // B col (N)

        // Unconditional LDS loads; lane-half handled with branch-free selects.
        const float rx = sx[r], ry = sy[r], rz = sz[r];
        const float nx = sx[n], ny = sy[n], nz = sz[n];

#if __has_builtin(__builtin_amdgcn_wmma_f32_16x16x4_f32)
        // 16x4 f32 A layout: lanes0-15 -> {K0,K1}, lanes16-31 -> {K2,K3=pad}
        v2f a, b;
        a.x = half ? rz : rx;   a.y = half ? 0.f : ry;
        b.x = half ? nz : nx;   b.y = half ? 0.f : ny;

        v8f acc = {};
        acc = __builtin_amdgcn_wmma_f32_16x16x4_f32(
            false, a, false, b, (short)0, acc, false, false);

        // D layout: VGPR v, lanes0-15 -> (M=v, N=m); lanes16-31 -> (M=v+8, N=m)
#pragma unroll
        for (int v = 0; v < 8; ++v)
            G[(ti * 16 + half * 8 + v) * GP + tj * 16 + m] = acc[v];
#else
#pragma unroll
        for (int v = 0; v < 8; ++v) {
            const int row = ti * 16 + half * 8 + v;
            const int col = tj * 16 + m;
            G[row * GP + col] = sx[row]*sx[col] + sy[row]*sy[col] + sz[row]*sz[col];
        }
#endif
    }
    __syncthreads();

    // ---- Feature sweep: coalesced stores of 1367 f32 per frame ----
    const size_t obase = (size_t)frame * NF;
    for (int f = tid; f < NF; f += BLK) {
        float val;
        if (f < ND) {
            // pairs ordered by separation s then start i; B(s)=(s-1)(100-s)/2 pairs precede s
            int s = (int)((101.0f - fast_sqrt((float)(9801 - 8 * f))) * 0.5f);
            int base = ((s - 1) * (100 - s)) >> 1;
            while (f < base)            { --s; base = ((s - 1) * (100 - s)) >> 1; }
            while (f >= base + NB - s)  { base += NB - s; ++s; }
            const int i = f - base;
            const int j = i + s;
            const float d2 = G[i * GP + i] + G[j * GP + j] - 2.f * G[i * GP + j];
            val = fast_sqrt(fmaxf(d2, 0.f));
        } else if (f < ND + NA) {
            const int t = f - ND;     // beads (t, t+1, t+2), vertex t+1
            const float b1x = sx[t] - sx[t+1], b1y = sy[t] - sy[t+1], b1z = sz[t] - sz[t+1];
            const float b2x = sx[t+2] - sx[t+1], b2y = sy[t+2] - sy[t+1], b2z = sz[t+2] - sz[t+1];
            const float dot = b1x*b2x + b1y*b2y + b1z*b2z;
            const float n1  = fast_sqrt(b1x*b1x + b1y*b1y + b1z*b1z);
            const float n2  = fast_sqrt(b2x*b2x + b2y*b2y + b2z*b2z);
            float ct = dot / (n1 * n2);
            ct = fminf(1.f, fmaxf(-1.f, ct));
            val = acosf(ct);
        } else {
            int t, want_sin;
            if (f < ND + NA + NDH) { t = f - (ND + NA);       want_sin = 0; }
            else                   { t = f - (ND + NA + NDH); want_sin = 1; }
            // beads t..t+3
            const float ax = sx[t+1]-sx[t],   ay = sy[t+1]-sy[t],   az = sz[t+1]-sz[t];
            const float bx = sx[t+2]-sx[t+1], by = sy[t+2]-sy[t+1], bz = sz[t+2]-sz[t+1];
            const float cx = sx[t+3]-sx[t+2], cy = sy[t+3]-sy[t+2], cz = sz[t+3]-sz[t+2];
            // p1 = a x b ; p2 = b x c
            const float p1x = ay*bz - az*by, p1y = az*bx - ax*bz, p1z = ax*by - ay*bx;
            const float p2x = by*cz - bz*cy, p2y = bz*cx - bx*cz, p2z = bx*cy - by*cx;
            const float n1 = fast_sqrt(p1x*p1x + p1y*p1y + p1z*p1z);
            const float n2 = fast_sqrt(p2x*p2x + p2y*p2y + p2z*p2z);
            if (!want_sin) {
                val = (p1x*p2x + p1y*p2y + p1z*p2z) / (n1 * n2);
            } else {
                const float qx = p1y*p2z - p1z*p2y;
                const float qy = p1z*p2x - p1x*p2z;
                const float qz = p1x*p2y - p1y*p2x;
                const float nb = fast_sqrt(bx*bx + by*by + bz*bz);
                val = (qx*bx + qy*by + qz*bz) / (n1 * n2 * nb);
            }
        }
        out[obase + f] = val;
    }
}

extern "C" void kernel_launch(void* const* d_in, const int* in_sizes, int n_in,
                              void* d_out, int out_size, void* d_ws, size_t ws_size,
                              hipStream_t stream) {
    const float* data = (const float*)d_in[0];
    float* out = (float*)d_out;
    const int n_frames = in_sizes[0] / (NB * 3);
    backbone_feat_kernel<<<dim3(n_frames), dim3(BLK), 0, stream>>>(data, out, n_frames);
}